// HCNLayer_58153857188495
// MI455X (gfx1250) — compile-verified
//
#include <hip/hip_runtime.h>
#include <hip/hip_bf16.h>

typedef __attribute__((ext_vector_type(2))) float v2f;
typedef __attribute__((ext_vector_type(8))) float v8f;

#define DIM 256
#define TM  16   // node rows per block
#define BN_EPS 1e-5f

__device__ inline v8f wmma_f32(v2f a, v2f b, v8f c) {
    // D = A(16x4) * B(4x16) + C, fp32 matrix pipe
    return __builtin_amdgcn_wmma_f32_16x16x4_f32(
        /*neg_a=*/false, a, /*neg_b=*/false, b,
        /*c_mod=*/(short)0, c, /*reuse_a=*/false, /*reuse_b=*/false);
}

// Compute two 16x16 output tiles (cols [ct0*16, ct0*16+32)) of  sA(16x256) @ W^T
// where W is row-major [256 out][256 in]. Accumulates into c0/c1.
__device__ inline void gemm_tiles(const float* __restrict__ sA,
                                  const float* __restrict__ W,
                                  int lane, int ct0, v8f& c0, v8f& c1) {
    const int mr   = lane & 15;
    const int koff = (lane >> 4) * 2;          // 0 or 2
    const float* arow = sA + mr * DIM + koff;
    const float* wr0  = W + (size_t)((ct0 * 16)      + mr) * DIM + koff;
    const float* wr1  = W + (size_t)((ct0 * 16 + 16) + mr) * DIM + koff;
#pragma unroll 4
    for (int k = 0; k < DIM; k += 4) {
        v2f a  = *(const v2f*)(arow + k);
        v2f b0 = *(const v2f*)(wr0 + k);
        v2f b1 = *(const v2f*)(wr1 + k);
        c0 = wmma_f32(a, b0, c0);
        c1 = wmma_f32(a, b1, c1);
    }
}

// ---------------- zero workspace ----------------
__global__ void zero_f4(float4* p, long long n4) {
    long long i = (long long)blockIdx.x * blockDim.x + threadIdx.x;
    long long stride = (long long)gridDim.x * blockDim.x;
    float4 z = make_float4(0.f, 0.f, 0.f, 0.f);
    for (; i < n4; i += stride) p[i] = z;
}

// ---------------- GINE edge scatter: aggr1[dst] += relu(x[src]+e) ------------
__global__ __launch_bounds__(256) void gine_edges(
    const float* __restrict__ x, const int* __restrict__ ei,
    const float* __restrict__ eattr, float* __restrict__ aggr, int E) {
    int e = blockIdx.x * 8 + (threadIdx.x >> 5);
    if (e >= E) return;
    int lane = threadIdx.x & 31;
    int src = ei[e];
    int dst = ei[E + e];
    const float* xs = x + (size_t)src * DIM;
    const float* ea = eattr + (size_t)e * DIM;
    float* ag = aggr + (size_t)dst * DIM;
#pragma unroll
    for (int j = 0; j < 8; ++j) {
        int c = lane + 32 * j;
        float v = fmaxf(xs[c] + ea[c], 0.0f);
        unsafeAtomicAdd(&ag[c], v);   // global_atomic_add_f32 (no return)
    }
}

// ---------------- GraphConv edge scatter: aggr2[dst] += x[src] ---------------
__global__ __launch_bounds__(256) void gconv_edges(
    const float* __restrict__ x, const int* __restrict__ ei,
    float* __restrict__ aggr, int E) {
    int e = blockIdx.x * 8 + (threadIdx.x >> 5);
    if (e >= E) return;
    int lane = threadIdx.x & 31;
    int src = ei[e];
    int dst = ei[E + e];
    const float* xs = x + (size_t)src * DIM;
    float* ag = aggr + (size_t)dst * DIM;
#pragma unroll
    for (int j = 0; j < 8; ++j) {
        int c = lane + 32 * j;
        unsafeAtomicAdd(&ag[c], xs[c]);
    }
}

// ---------------- fused WMMA kernel: 16 rows per block, 8 waves --------------
__global__ __launch_bounds__(256) void hcn_fused(
    const float* __restrict__ x, const float* __restrict__ aggr1,
    const float* __restrict__ aggr2, const float* __restrict__ d_eps,
    const float* __restrict__ w1a, const float* __restrict__ b1a,
    const float* __restrict__ bn1g, const float* __restrict__ bn1b,
    const float* __restrict__ bn1m, const float* __restrict__ bn1v,
    const float* __restrict__ w1b, const float* __restrict__ b1b,
    const float* __restrict__ wrel, const float* __restrict__ brel,
    const float* __restrict__ wroot,
    const float* __restrict__ d_a1, const float* __restrict__ d_a2,
    const float* __restrict__ bng, const float* __restrict__ bnb,
    const float* __restrict__ bnm, const float* __restrict__ bnv,
    float* __restrict__ out, int n_nodes) {
    __shared__ float s_hin[TM * DIM];   // GINE pre-MLP input; reused for t
    __shared__ float s_x[TM * DIM];
    __shared__ float s_a2[TM * DIM];

    const int tid  = threadIdx.x;
    const int lane = tid & 31;
    const int wave = tid >> 5;
    const int ct0  = wave * 2;          // this wave's first of 2 column tiles
    const int mr   = lane & 15;
    const int hi   = lane >> 4;

    const long long base  = (long long)blockIdx.x * TM * DIM;
    const long long total = (long long)n_nodes * DIM;
    const float eps1 = 1.0f + d_eps[0];
    const float a1 = d_a1[0], a2 = d_a2[0];

    // ---- stage x, aggr2, h_in = (1+eps)*x + aggr1 (contiguous float4 copy)
#pragma unroll
    for (int i = 0; i < 4; ++i) {
        int idx = i * 1024 + tid * 4;
        long long g = base + idx;
        float4 xv = make_float4(0.f, 0.f, 0.f, 0.f), av = xv, bv = xv;
        if (g < total) {
            xv = *(const float4*)(x + g);
            av = *(const float4*)(aggr1 + g);
            bv = *(const float4*)(aggr2 + g);
        }
        *(float4*)(s_x + idx)  = xv;
        *(float4*)(s_a2 + idx) = bv;
        float4 h;
        h.x = fmaf(eps1, xv.x, av.x);
        h.y = fmaf(eps1, xv.y, av.y);
        h.z = fmaf(eps1, xv.z, av.z);
        h.w = fmaf(eps1, xv.w, av.w);
        *(float4*)(s_hin + idx) = h;
    }
    __syncthreads();

    // ---- GEMM1: h_in @ w1a^T
    v8f c0 = {}, c1 = {};
    gemm_tiles(s_hin, w1a, lane, ct0, c0, c1);
    __syncthreads();   // everyone done reading s_hin before we overwrite with t

    // ---- epilogue 1: bias + BN(eval) + ReLU -> t (into s_hin)
#pragma unroll
    for (int t = 0; t < 2; ++t) {
        int n = (ct0 + t) * 16 + mr;
        float sc = bn1g[n] / sqrtf(bn1v[n] + BN_EPS);
        float sh = bn1b[n] - bn1m[n] * sc;
        float bb = b1a[n];
        v8f& c = t ? c1 : c0;
#pragma unroll
        for (int r = 0; r < 8; ++r) {
            float v = (c[r] + bb) * sc + sh;
            s_hin[(r + 8 * hi) * DIM + n] = fmaxf(v, 0.0f);
        }
    }
    __syncthreads();

    // ---- GEMM2: h_down = t @ w1b^T
    v8f d0 = {}, d1 = {};
    gemm_tiles(s_hin, w1b, lane, ct0, d0, d1);

    // ---- GEMM3+4 (chained accumulate): h_up = aggr2 @ wrel^T + x @ wroot^T
    v8f e0 = {}, e1 = {};
    gemm_tiles(s_a2, wrel, lane, ct0, e0, e1);
    gemm_tiles(s_x, wroot, lane, ct0, e0, e1);

    // ---- final: relu(BN(x + a1*(h_down + b1b) + a2*(h_up + brel)))
#pragma unroll
    for (int t = 0; t < 2; ++t) {
        int n = (ct0 + t) * 16 + mr;
        float sc = bng[n] / sqrtf(bnv[n] + BN_EPS);
        float sh = bnb[n] - bnm[n] * sc;
        float bd = b1b[n];
        float br = brel[n];
        v8f& d = t ? d1 : d0;
        v8f& e = t ? e1 : e0;
#pragma unroll
        for (int r = 0; r < 8; ++r) {
            int row = r + 8 * hi;
            float xv = s_x[row * DIM + n];
            float o = xv + a1 * (d[r] + bd) + a2 * (e[r] + br);
            o = fmaxf(o * sc + sh, 0.0f);
            long long g = base + (long long)row * DIM + n;
            if (g < total) out[g] = o;
        }
    }
}

extern "C" void kernel_launch(void* const* d_in, const int* in_sizes, int n_in,
                              void* d_out, int out_size, void* d_ws, size_t ws_size,
                              hipStream_t stream) {
    const float* x     = (const float*)d_in[0];
    const int*   ei    = (const int*)d_in[1];
    const float* eattr = (const float*)d_in[2];
    const int*   vi    = (const int*)d_in[3];
    const float* eps   = (const float*)d_in[4];
    const float* w1a   = (const float*)d_in[5];
    const float* b1a   = (const float*)d_in[6];
    const float* bn1g  = (const float*)d_in[7];
    const float* bn1b  = (const float*)d_in[8];
    const float* bn1m  = (const float*)d_in[9];
    const float* bn1v  = (const float*)d_in[10];
    const float* w1b   = (const float*)d_in[11];
    const float* b1b   = (const float*)d_in[12];
    const float* wrel  = (const float*)d_in[13];
    const float* brel  = (const float*)d_in[14];
    const float* wroot = (const float*)d_in[15];
    const float* a1    = (const float*)d_in[16];
    const float* a2    = (const float*)d_in[17];
    const float* bng   = (const float*)d_in[18];
    const float* bnb   = (const float*)d_in[19];
    const float* bnm   = (const float*)d_in[20];
    const float* bnv   = (const float*)d_in[21];
    float* out = (float*)d_out;

    const int N  = in_sizes[0] / DIM;
    const int E1 = in_sizes[1] / 2;
    const int E2 = in_sizes[3] / 2;

    float* aggr1 = (float*)d_ws;
    float* aggr2 = aggr1 + (size_t)N * DIM;

    long long n4 = (long long)N * DIM * 2 / 4;
    zero_f4<<<2048, 256, 0, stream>>>((float4*)d_ws, n4);
    gine_edges<<<(E1 + 7) / 8, 256, 0, stream>>>(x, ei, eattr, aggr1, E1);
    gconv_edges<<<(E2 + 7) / 8, 256, 0, stream>>>(x, vi, aggr2, E2);
    hcn_fused<<<(N + TM - 1) / TM, 256, 0, stream>>>(
        x, aggr1, aggr2, eps, w1a, b1a, bn1g, bn1b, bn1m, bn1v,
        w1b, b1b, wrel, brel, wroot, a1, a2, bng, bnb, bnm, bnv, out, N);
}